// MoeLayer_29291676959122
// MI455X (gfx1250) — compile-verified
//
#include <hip/hip_runtime.h>
#include <math.h>

#define T_TOK 2048
#define D_DIM 1024
#define F_DIM 2048
#define N_EXP 8
#define O_DIM 9
#define ALPHA_C 0.01f

// workspace layout (byte offsets)
#define WS_WINNER   0u           // int[2048]
#define WS_COUNTS   8192u        // int[8]
#define WS_OFFSETS  8224u        // int[9]
#define WS_CURSORS  8260u        // int[8]
#define WS_TOKIDS   8292u        // int[2048]
#define WS_ZPART    16512u       // float[256]
#define WS_PPART    17536u       // float[2048]
#define WS_FPART    25728u       // int[2048]
#define WS_H        65536u       // float[2048*2048] = 16 MB

typedef __attribute__((ext_vector_type(2))) float v2f;
typedef __attribute__((ext_vector_type(8))) float v8f;

// ---------------------------------------------------------------------------
// Kernel 1: gate GEMM (T x D @ D x 8), winner selection, loss partials.
// One wave per token; deterministic per-block tree reductions (no fp atomics).
// ---------------------------------------------------------------------------
__global__ __launch_bounds__(256) void gate_kernel(
    const float* __restrict__ x, const float* __restrict__ gw,
    const float* __restrict__ gb, int* __restrict__ winner,
    float* __restrict__ z_part, float* __restrict__ p_part,
    int* __restrict__ f_part)
{
    __shared__ float s_z[8];
    __shared__ float s_p[8][8];
    __shared__ int   s_a[8];

    int lane = threadIdx.x & 31;
    int wid  = threadIdx.x >> 5;
    int t    = blockIdx.x * 8 + wid;

    float acc[8];
#pragma unroll
    for (int n = 0; n < 8; ++n) acc[n] = 0.f;

    const float* xp = x + (size_t)t * D_DIM;
    for (int i = 0; i < D_DIM / 32; ++i) {
        int d = i * 32 + lane;
        float xv = xp[d];               // coalesced across lanes
        const float* g = gw + d * 8;
#pragma unroll
        for (int n = 0; n < 8; ++n) acc[n] += xv * g[n];
    }
#pragma unroll
    for (int off = 16; off >= 1; off >>= 1)
#pragma unroll
        for (int n = 0; n < 8; ++n) acc[n] += __shfl_down(acc[n], off, 32);

    if (lane == 0) {
        float lg[8];
        float mx = -1e30f;
#pragma unroll
        for (int n = 0; n < 8; ++n) { lg[n] = acc[n] + gb[n]; mx = fmaxf(mx, lg[n]); }
        float pr[8], se = 0.f;
#pragma unroll
        for (int n = 0; n < 8; ++n) { pr[n] = expf(lg[n] - mx); se += pr[n]; }
        float inv = 1.f / se;
#pragma unroll
        for (int n = 0; n < 8; ++n) pr[n] *= inv;
        float lse = mx + logf(se);

        // argmax (first occurrence, matches jnp.argmax)
        int i1 = 0; float v1 = lg[0];
#pragma unroll
        for (int n = 1; n < 8; ++n) if (lg[n] > v1) { v1 = lg[n]; i1 = n; }
        // second-best index (first occurrence among n != i1, matches top_k)
        int i2 = -1; float v2 = -1e30f;
#pragma unroll
        for (int n = 0; n < 8; ++n) if (n != i1 && lg[n] > v2) { v2 = lg[n]; i2 = n; }
        int w = i1 > i2 ? i1 : i2;      // last-writer-wins expert id

        winner[t] = w;
        s_z[wid] = lse * lse;
#pragma unroll
        for (int n = 0; n < 8; ++n) s_p[wid][n] = pr[n];
        s_a[wid] = i1;
    }
    __syncthreads();
    if (threadIdx.x == 0) {
        float z = 0.f, p[8]; int fc[8];
        for (int n = 0; n < 8; ++n) { p[n] = 0.f; fc[n] = 0; }
        for (int w = 0; w < 8; ++w) {
            z += s_z[w];
            for (int n = 0; n < 8; ++n) p[n] += s_p[w][n];
            fc[s_a[w]] += 1;
        }
        z_part[blockIdx.x] = z;
        for (int n = 0; n < 8; ++n) {
            p_part[blockIdx.x * 8 + n] = p[n];
            f_part[blockIdx.x * 8 + n] = fc[n];
        }
    }
}

// ---------------------------------------------------------------------------
// Kernel 2: deterministic final reduction of losses + bucket offsets.
// ---------------------------------------------------------------------------
__global__ void finalize_kernel(
    const float* __restrict__ z_part, const float* __restrict__ p_part,
    const int* __restrict__ f_part, int* __restrict__ counts,
    int* __restrict__ offsets, int* __restrict__ cursors,
    float* __restrict__ out_losses)
{
    if (threadIdx.x != 0 || blockIdx.x != 0) return;
    float z = 0.f, p[8]; int fc[8];
    for (int n = 0; n < 8; ++n) { p[n] = 0.f; fc[n] = 0; }
    for (int b = 0; b < 256; ++b) {
        z += z_part[b];
        for (int n = 0; n < 8; ++n) { p[n] += p_part[b * 8 + n]; fc[n] += f_part[b * 8 + n]; }
    }
    out_losses[0] = z / (float)T_TOK;                      // loss_z
    float lb = 0.f;
    for (int n = 0; n < 8; ++n)
        lb += ((float)fc[n] / (float)T_TOK) * (p[n] / (float)T_TOK);
    out_losses[1] = lb * ALPHA_C * (float)N_EXP;           // loss_b
    int acc = 0;
    for (int n = 0; n < 8; ++n) { counts[n] = fc[n]; offsets[n] = acc; acc += fc[n]; cursors[n] = 0; }
    offsets[8] = acc;
}

// ---------------------------------------------------------------------------
// Kernel 3: bucket token ids by winner expert (int atomics only).
// ---------------------------------------------------------------------------
__global__ __launch_bounds__(256) void scatter_kernel(
    const int* __restrict__ winner, const int* __restrict__ offsets,
    int* __restrict__ cursors, int* __restrict__ tokids)
{
    int t = blockIdx.x * 256 + threadIdx.x;
    int e = winner[t];
    int pos = offsets[e] + atomicAdd(&cursors[e], 1);
    tokids[pos] = t;
}

// ---------------------------------------------------------------------------
// Kernel 4: grouped expert GEMM via V_WMMA_F32_16X16X4_F32 + fused bias/ReLU.
// Block = 256 threads (8 waves). blockIdx.x = 32-token tile, blockIdx.y = F/256.
// Each wave owns a 32(token) x 32(F) output = 4 accumulator tiles, K = 1024.
// Per K-step of 4: 2 ds_load_b64 (A) + 4 global_load_b32 (B, immediate offsets)
// feed 4 WMMAs -> 1.5 mem-ops per WMMA.
// ---------------------------------------------------------------------------
#define M_TILE 32
#define KCHUNK 256
#define AS_STRIDE 268   // 268*4B rows: 16B-aligned, m*268 % 64 distinct -> conflict-free b64

__global__ __launch_bounds__(256) void ffn1_kernel(
    const float* __restrict__ x, const float* __restrict__ w1,
    const float* __restrict__ b1, const int* __restrict__ counts,
    const int* __restrict__ offsets, const int* __restrict__ tokids,
    float* __restrict__ h)
{
    __shared__ float As[M_TILE * AS_STRIDE];   // 34,304 B
    __shared__ int toks[M_TILE];

    // map linear tile id -> (expert, tile-within-expert)
    int tile = blockIdx.x;
    int e = -1, t_in = 0;
    {
        int accum = 0;
        for (int i = 0; i < N_EXP; ++i) {
            int nt = (counts[i] + M_TILE - 1) / M_TILE;
            if (e < 0 && tile < accum + nt) { e = i; t_in = tile - accum; }
            accum += nt;
        }
    }
    if (e < 0) return;                      // uniform across block
    int rowsStart = offsets[e] + t_in * M_TILE;
    int rowsCount = counts[e] - t_in * M_TILE;
    if (rowsCount > M_TILE) rowsCount = M_TILE;

    int tid = threadIdx.x;
    if (tid < M_TILE) toks[tid] = (tid < rowsCount) ? tokids[rowsStart + tid] : 0;

    int lane = tid & 31;
    int wave = tid >> 5;
    int half = lane >> 4;                   // 0: K+0/K+1 ; 1: K+2/K+3
    int n    = lane & 15;
    int fA   = blockIdx.y * 256 + wave * 32 + n;   // second F tile at fA+16

    const float* wbase = w1 + (size_t)e * D_DIM * F_DIM;

    v8f acc00 = {}, acc01 = {}, acc10 = {}, acc11 = {};

    for (int kc = 0; kc < D_DIM; kc += KCHUNK) {
        __syncthreads();                    // prior-iter LDS reads done
        // stage A chunk: 32 rows x 256 K (gathered, zero-padded rows)
        for (int idx = tid; idx < M_TILE * (KCHUNK / 4); idx += 256) {
            int row = idx >> 6;             // 64 float4 per row
            int c4  = idx & 63;
            float4 v = make_float4(0.f, 0.f, 0.f, 0.f);
            if (row < rowsCount) {
                const float* src = x + (size_t)toks[row] * D_DIM + kc + c4 * 4;
                v = *(const float4*)src;
            }
            *(float4*)(&As[row * AS_STRIDE + c4 * 4]) = v;
        }
        __syncthreads();

        const float* bp = wbase + (size_t)(kc + half * 2) * F_DIM + fA;
        const float* ap = &As[(lane & 15) * AS_STRIDE + half * 2];
#pragma unroll 4
        for (int k = 0; k < KCHUNK; k += 4) {
            v2f a0 = *(const v2f*)(ap + k);                     // rows 0..15
            v2f a1 = *(const v2f*)(ap + 16 * AS_STRIDE + k);    // rows 16..31
            v2f b0, b1;
            b0.x = bp[0];            // w1[e][k+kb  ][fA]
            b0.y = bp[F_DIM];        // w1[e][k+kb+1][fA]
            b1.x = bp[16];           // w1[e][k+kb  ][fA+16]
            b1.y = bp[F_DIM + 16];   // w1[e][k+kb+1][fA+16]
            bp += 4 * F_DIM;
            acc00 = __builtin_amdgcn_wmma_f32_16x16x4_f32(
                false, a0, false, b0, (short)0, acc00, false, false);
            acc01 = __builtin_amdgcn_wmma_f32_16x16x4_f32(
                false, a0, false, b1, (short)0, acc01, false, false);
            acc10 = __builtin_amdgcn_wmma_f32_16x16x4_f32(
                false, a1, false, b0, (short)0, acc10, false, false);
            acc11 = __builtin_amdgcn_wmma_f32_16x16x4_f32(
                false, a1, false, b1, (short)0, acc11, false, false);
        }
    }

    // epilogue: + b1, ReLU, scatter h rows for valid tokens
    float biasA = b1[(size_t)e * F_DIM + fA];
    float biasB = b1[(size_t)e * F_DIM + fA + 16];
#pragma unroll
    for (int r = 0; r < 8; ++r) {
        int m = r + half * 8;               // C layout: VGPR r -> M=r / M=r+8 per lane half
        if (m < rowsCount) {
            int tok = toks[m];
            h[(size_t)tok * F_DIM + fA]      = fmaxf(acc00[r] + biasA, 0.f);
            h[(size_t)tok * F_DIM + fA + 16] = fmaxf(acc01[r] + biasB, 0.f);
        }
        int m2 = m + 16;
        if (m2 < rowsCount) {
            int tok = toks[m2];
            h[(size_t)tok * F_DIM + fA]      = fmaxf(acc10[r] + biasA, 0.f);
            h[(size_t)tok * F_DIM + fA + 16] = fmaxf(acc11[r] + biasB, 0.f);
        }
    }
}

// ---------------------------------------------------------------------------
// Kernel 5: y[t] = h[t] @ w2[winner[t]] + b2[winner[t]]  (F=2048 -> O=9)
// One wave per token, deterministic shuffle reduction.
// ---------------------------------------------------------------------------
__global__ __launch_bounds__(256) void ffn2_kernel(
    const float* __restrict__ h, const float* __restrict__ w2,
    const float* __restrict__ b2, const int* __restrict__ winner,
    float* __restrict__ out)
{
    int lane = threadIdx.x & 31;
    int wid  = threadIdx.x >> 5;
    int t    = blockIdx.x * 8 + wid;
    int e    = winner[t];
    const float* hp = h + (size_t)t * F_DIM;
    const float* wp = w2 + (size_t)e * F_DIM * O_DIM;

    float acc[O_DIM];
#pragma unroll
    for (int o = 0; o < O_DIM; ++o) acc[o] = 0.f;

    for (int fidx = lane; fidx < F_DIM; fidx += 32) {
        float hv = hp[fidx];                // coalesced
        const float* wr = wp + fidx * O_DIM;
#pragma unroll
        for (int o = 0; o < O_DIM; ++o) acc[o] += hv * wr[o];
    }
#pragma unroll
    for (int off = 16; off >= 1; off >>= 1)
#pragma unroll
        for (int o = 0; o < O_DIM; ++o) acc[o] += __shfl_down(acc[o], off, 32);

    if (lane == 0) {
#pragma unroll
        for (int o = 0; o < O_DIM; ++o)
            out[(size_t)t * O_DIM + o] = acc[o] + b2[e * O_DIM + o];
    }
}

// ---------------------------------------------------------------------------
extern "C" void kernel_launch(void* const* d_in, const int* in_sizes, int n_in,
                              void* d_out, int out_size, void* d_ws, size_t ws_size,
                              hipStream_t stream)
{
    const float* x  = (const float*)d_in[0];   // [2048,1024]
    const float* gw = (const float*)d_in[1];   // [1024,8]
    const float* gb = (const float*)d_in[2];   // [8]
    const float* w1 = (const float*)d_in[3];   // [8,1024,2048]
    const float* b1 = (const float*)d_in[4];   // [8,2048]
    const float* w2 = (const float*)d_in[5];   // [8,2048,9]
    const float* b2 = (const float*)d_in[6];   // [8,9]
    float* out = (float*)d_out;                // [2048*9] ++ [loss_z, loss_b]

    char* ws = (char*)d_ws;
    int*   winner  = (int*)(ws + WS_WINNER);
    int*   counts  = (int*)(ws + WS_COUNTS);
    int*   offsets = (int*)(ws + WS_OFFSETS);
    int*   cursors = (int*)(ws + WS_CURSORS);
    int*   tokids  = (int*)(ws + WS_TOKIDS);
    float* z_part  = (float*)(ws + WS_ZPART);
    float* p_part  = (float*)(ws + WS_PPART);
    int*   f_part  = (int*)(ws + WS_FPART);
    float* h       = (float*)(ws + WS_H);

    gate_kernel<<<256, 256, 0, stream>>>(x, gw, gb, winner, z_part, p_part, f_part);
    finalize_kernel<<<1, 64, 0, stream>>>(z_part, p_part, f_part, counts, offsets,
                                          cursors, out + (size_t)T_TOK * O_DIM);
    scatter_kernel<<<8, 256, 0, stream>>>(winner, offsets, cursors, tokids);
    // max 32-token tiles = sum_e ceil(count_e/32) <= 71; grid.x=72, extras early-exit
    dim3 g4(72, 8);
    ffn1_kernel<<<g4, 256, 0, stream>>>(x, w1, b1, counts, offsets, tokids, h);
    ffn2_kernel<<<256, 256, 0, stream>>>(h, w2, b2, winner, out);
}